// GeometricAssigner_67997922230571
// MI455X (gfx1250) — compile-verified
//
#include <hip/hip_runtime.h>
#include <hip/hip_bf16.h>
#include <stdint.h>

// GeometricAssigner: for each edge, nearest of 27 grid offsets {-v,0,v}^3 to
// (ref_xyz[e_ref] - query_xyz[e_query]).  The 3x3x3 grid is separable, so the
// 27-way argmin is 2 threshold compares per axis (ties match jnp.argmin's
// first-index rule).  Memory-bound: ~132 MB HBM => ~5.7us floor at 23.3 TB/s.
// Index streams staged via CDNA5 async global->LDS B128 copies (ASYNCcnt),
// double-buffered per wave; gathers stay L2-resident (32 MB tables << 192 MB).

#define BLOCK 256
#define WAVES (BLOCK / 32)
#define CHUNK 64            // edges per wave per iteration (2 per lane)
#define EDGES_PER_THREAD 16

__global__ __launch_bounds__(BLOCK)
void geo_assign_kernel(const float4* __restrict__ ref4,
                       const float4* __restrict__ qry4,
                       const long long* __restrict__ e_ref,
                       const long long* __restrict__ e_qry,
                       const float* __restrict__ kp,
                       long long* __restrict__ out,
                       long long E)
{
    // [wave][buf][arr(ref/qry)][edge-in-chunk]
    __shared__ long long sbuf[WAVES][2][2][CHUNK];

    const int lane = threadIdx.x & 31;
    const int wave = threadIdx.x >> 5;

    // kernel_pos[26] = (+vx,+vy,+vz); decision thresholds are +-v/2.
    const float tx = 0.5f * kp[78];
    const float ty = 0.5f * kp[79];
    const float tz = 0.5f * kp[80];

    const long long stride = (long long)gridDim.x * BLOCK * 2;
    long long base = ((long long)blockIdx.x * WAVES + wave) * CHUNK;

    // LDS byte addresses (low 32 bits of the generic shared-aperture address).
    uint32_t lds_r[2], lds_q[2];
    lds_r[0] = (uint32_t)(uintptr_t)&sbuf[wave][0][0][2 * lane];
    lds_q[0] = (uint32_t)(uintptr_t)&sbuf[wave][0][1][2 * lane];
    lds_r[1] = (uint32_t)(uintptr_t)&sbuf[wave][1][0][2 * lane];
    lds_q[1] = (uint32_t)(uintptr_t)&sbuf[wave][1][1][2 * lane];

    // Async-stage one full 64-edge chunk (2 int64 per lane per stream) into
    // LDS buffer b.  Called only under a wave-uniform "chunk fully in range"
    // condition, so each call is exactly +2 on ASYNCcnt.
    auto stage = [&](int b, long long cbase) {
        unsigned long long ar = (unsigned long long)(uintptr_t)(e_ref + cbase + 2 * lane);
        unsigned long long aq = (unsigned long long)(uintptr_t)(e_qry + cbase + 2 * lane);
        asm volatile("global_load_async_to_lds_b128 %0, %1, off"
                     :: "v"(lds_r[b]), "v"(ar) : "memory");
        asm volatile("global_load_async_to_lds_b128 %0, %1, off"
                     :: "v"(lds_q[b]), "v"(aq) : "memory");
    };

    // Separable nearest-of-{-v,0,v} per axis.
    auto assign = [&](long long ir, long long iq) -> long long {
        const float4 r = ref4[(int)ir];   // global_load_b128 gather (L2-resident)
        const float4 q = qry4[(int)iq];
        const float rx = r.y - q.y;
        const float ry = r.z - q.z;
        const float rz = r.w - q.w;
        const int ix = (int)(rx > -tx) + (int)(rx > tx);
        const int iy = (int)(ry > -ty) + (int)(ry > ty);
        const int iz = (int)(rz > -tz) + (int)(rz > tz);
        return (long long)(ix * 9 + iy * 3 + iz);
    };

    int buf = 0;
    bool cur_staged = (base + CHUNK <= E);
    if (cur_staged) stage(0, base);

    for (; base < E; base += stride) {
        const long long nbase = base + stride;
        const bool nxt_staged = (nbase + CHUNK <= E);
        if (nxt_staged) {
            stage(buf ^ 1, nbase);                            // 2 more in flight
            asm volatile("s_wait_asynccnt 0x2" ::: "memory"); // current chunk done
        } else if (cur_staged) {
            asm volatile("s_wait_asynccnt 0x0" ::: "memory");
        }

        const long long i0 = base + 2 * lane;
        if (cur_staged) {
            const long long ir0 = sbuf[wave][buf][0][2 * lane];
            const long long ir1 = sbuf[wave][buf][0][2 * lane + 1];
            const long long iq0 = sbuf[wave][buf][1][2 * lane];
            const long long iq1 = sbuf[wave][buf][1][2 * lane + 1];
            // WAR guard: LDS reads must land before the async engine may
            // overwrite this buffer next iteration (async unordered vs DS).
            asm volatile("s_wait_dscnt 0x0" ::: "memory");

            longlong2 o;
            o.x = assign(ir0, iq0);
            o.y = assign(ir1, iq1);
            *(longlong2*)(out + i0) = o;                      // b128 store
        } else {
            // Tail chunk: direct guarded loads (no LDS involvement).
            if (i0 < E)     out[i0]     = assign(e_ref[i0],     e_qry[i0]);
            if (i0 + 1 < E) out[i0 + 1] = assign(e_ref[i0 + 1], e_qry[i0 + 1]);
        }
        cur_staged = nxt_staged;
        buf ^= 1;
    }
}

extern "C" void kernel_launch(void* const* d_in, const int* in_sizes, int n_in,
                              void* d_out, int out_size, void* d_ws, size_t ws_size,
                              hipStream_t stream) {
    const float4* ref4      = (const float4*)d_in[0];      // ref_bxyz   (N_REF,4) f32
    const float4* qry4      = (const float4*)d_in[1];      // query_bxyz (N_QRY,4) f32
    const long long* e_ref  = (const long long*)d_in[2];   // (E,) int64
    const long long* e_qry  = (const long long*)d_in[3];   // (E,) int64
    const float* kp         = (const float*)d_in[4];       // (27,3) f32
    long long* out          = (long long*)d_out;           // (E,) int64

    const long long E = (long long)in_sizes[2];
    if (E <= 0) return;

    const long long per_blk = (long long)BLOCK * EDGES_PER_THREAD;
    long long want = (E + per_blk - 1) / per_blk;
    if (want < 1) want = 1;
    if (want > 1048576) want = 1048576;
    const int blocks = (int)want;

    geo_assign_kernel<<<blocks, BLOCK, 0, stream>>>(ref4, qry4, e_ref, e_qry,
                                                    kp, out, E);
}